// NodeMixup_65893388255597
// MI455X (gfx1250) — compile-verified
//
#include <hip/hip_runtime.h>
#include <math.h>

// Problem constants (match reference)
#define NN   50000
#define EE   800000
#define CH   128
#define OUTC 64

typedef float v2f __attribute__((ext_vector_type(2)));
typedef float v8f __attribute__((ext_vector_type(8)));

__device__ __forceinline__ float4 ld4(const float* p){ return *(const float4*)p; }
__device__ __forceinline__ void   st4(float* p, float4 v){ *(float4*)p = v; }
__device__ __forceinline__ float4 relu4(float4 v){
    float4 r; r.x=fmaxf(v.x,0.f); r.y=fmaxf(v.y,0.f); r.z=fmaxf(v.z,0.f); r.w=fmaxf(v.w,0.f); return r;
}

// ---------------- small utility kernels ----------------

__global__ void fill_kernel(float* __restrict__ p, float v, int n){
    int i = blockIdx.x*blockDim.x + threadIdx.x;
    if (i < n) p[i] = v;
}

__global__ void count_deg_kernel(const int* __restrict__ dst, float* __restrict__ deg, int E){
    int i = blockIdx.x*blockDim.x + threadIdx.x;
    if (i < E) atomicAdd(&deg[dst[i]], 1.0f);
}

// in: dinv holds deg (=1+indeg). out: dinv = rsqrt(deg), deginv = 1/deg
__global__ void norms_kernel(float* __restrict__ dinv, float* __restrict__ deginv, int n){
    int i = blockIdx.x*blockDim.x + threadIdx.x;
    if (i < n){ float d = dinv[i]; dinv[i] = rsqrtf(d); deginv[i] = 1.0f/d; }
}

// x0 = x ; xmix = lam*x + (1-lam)*x[perm]
__global__ void init_mix_kernel(const float* __restrict__ x, const int* __restrict__ perm,
                                const float* __restrict__ lamp,
                                float* __restrict__ x0, float* __restrict__ xmix, int n){
    int t = blockIdx.x*blockDim.x + threadIdx.x;
    if (t >= n*32) return;
    int i = t >> 5, c = (t & 31) << 2;
    float lam = lamp[0];
    float4 a = ld4(x + (size_t)i*CH + c);
    float4 b = ld4(x + (size_t)perm[i]*CH + c);
    st4(x0 + (size_t)i*CH + c, a);
    float4 m; m.x = lam*a.x + (1.f-lam)*b.x; m.y = lam*a.y + (1.f-lam)*b.y;
              m.z = lam*a.z + (1.f-lam)*b.z; m.w = lam*a.w + (1.f-lam)*b.w;
    st4(xmix + (size_t)i*CH + c, m);
}

// agg[dst] += dinv[src]*dinv[dst] * Y[gather ? gather[src] : src]
// one wave per edge, float4 slice per lane -> 4 coalesced global_atomic_add_f32
__global__ void scatter_kernel(const float* __restrict__ Y,
                               const int* __restrict__ src, const int* __restrict__ dst,
                               const int* __restrict__ gather,
                               const float* __restrict__ dinv,
                               float* __restrict__ agg, int E){
    int t = blockIdx.x*blockDim.x + threadIdx.x;
    int e = t >> 5;
    if (e >= E) return;
    int lane = t & 31;
    int s = src[e], d = dst[e];
    float w = dinv[s] * dinv[d];
    int gs = gather ? gather[s] : s;
    float4 v = ld4(Y + (size_t)gs*CH + lane*4);
    float* ap = agg + (size_t)d*CH + lane*4;
    atomicAdd(ap+0, w*v.x);
    atomicAdd(ap+1, w*v.y);
    atomicAdd(ap+2, w*v.z);
    atomicAdd(ap+3, w*v.w);
}

// hout = lam*relu(agg + deginv*Ymix + b); if x0out: x0out = relu(agg + deginv*Y + b)
__global__ void combineA_kernel(const float* __restrict__ agg, const float* __restrict__ Y,
                                const float* __restrict__ Ymix, const float* __restrict__ deginv,
                                const float* __restrict__ bias, const float* __restrict__ lamp,
                                float* __restrict__ hout, float* __restrict__ x0out, int n){
    int t = blockIdx.x*blockDim.x + threadIdx.x;
    if (t >= n*32) return;
    int i = t >> 5, c = (t & 31) << 2;
    size_t o = (size_t)i*CH + c;
    float di = deginv[i], lam = lamp[0];
    float4 ag = ld4(agg + o), ym = ld4(Ymix + o), bv = ld4(bias + c);
    float4 h; h.x = ag.x + di*ym.x + bv.x; h.y = ag.y + di*ym.y + bv.y;
              h.z = ag.z + di*ym.z + bv.z; h.w = ag.w + di*ym.w + bv.w;
    h = relu4(h);
    float4 hh; hh.x = lam*h.x; hh.y = lam*h.y; hh.z = lam*h.z; hh.w = lam*h.w;
    st4(hout + o, hh);
    if (x0out){
        float4 y = ld4(Y + o);
        float4 x0n; x0n.x = ag.x + di*y.x + bv.x; x0n.y = ag.y + di*y.y + bv.y;
                    x0n.z = ag.z + di*y.z + bv.z; x0n.w = ag.w + di*y.w + bv.w;
        st4(x0out + o, relu4(x0n));
    }
}

// accum += (1-lam)*relu(agg + deginvb*Ymix + b)
__global__ void combineB_kernel(const float* __restrict__ agg, const float* __restrict__ Ymix,
                                const float* __restrict__ deginvb, const float* __restrict__ bias,
                                const float* __restrict__ lamp, float* __restrict__ accum, int n){
    int t = blockIdx.x*blockDim.x + threadIdx.x;
    if (t >= n*32) return;
    int i = t >> 5, c = (t & 31) << 2;
    size_t o = (size_t)i*CH + c;
    float di = deginvb[i], w = 1.0f - lamp[0];
    float4 ag = ld4(agg + o), ym = ld4(Ymix + o), bv = ld4(bias + c), acc = ld4(accum + o);
    float4 h; h.x = ag.x + di*ym.x + bv.x; h.y = ag.y + di*ym.y + bv.y;
              h.z = ag.z + di*ym.z + bv.z; h.w = ag.w + di*ym.w + bv.w;
    h = relu4(h);
    acc.x += w*h.x; acc.y += w*h.y; acc.z += w*h.z; acc.w += w*h.w;
    st4(accum + o, acc);
}

// ---------------- fp32 WMMA GEMM ----------------
// Y0 = X0 @ W (+bias), and optionally Y1 = X1 @ W (+bias) reusing the same LDS-staged W.
// W staged K-pair interleaved: (k,n) -> wlds[(k>>1)*PSTR + 2n + (k&1)], so the B operand
// {W[kk][n], W[kk+1][n]} is one aligned ds_load_b64 directly into the WMMA source pair.
// PSTR % 64 == 32 -> lanes 0-15 (pair-row p) and 16-31 (pair-row p+1) cover disjoint
// dword-bank halves: conflict-free 64-bit LDS reads.
template<int NT>
__global__ __launch_bounds__(256)
void gemm_wmma_f32(const float* __restrict__ X0v, float* __restrict__ Y0v,
                   const float* __restrict__ X1v, float* __restrict__ Y1v,
                   const float* __restrict__ W, const float* __restrict__ bias, int M){
    constexpr int K    = CH;            // 128
    constexpr int NC   = NT * 16;
    constexpr int PSTR = 2*NC + 32;     // dwords per K-pair row; 288 (NT=8) / 160 (NT=4)
    extern __shared__ float wlds[];

    int tid = threadIdx.x;
    for (int idx = tid; idx < K*NC; idx += 256){
        int r = idx / NC, c = idx - r*NC;
        wlds[(r >> 1)*PSTR + c*2 + (r & 1)] = W[idx];
    }
    __syncthreads();

    int wave = tid >> 5;
    int lane = tid & 31;
    int ln   = lane & 15;
    int half = lane >> 4;                 // A/B: +2 in K; C/D: +8 in M
    int row0 = (blockIdx.x*8 + wave) * 16;
    int rA   = row0 + ln; if (rA > M-1) rA = M-1;   // clamp loads; EXEC all-ones for WMMA

    for (int s = 0; s < 2; ++s){
        const float* X = s ? X1v : X0v;
        float*       Y = s ? Y1v : Y0v;
        if (!X) break;                    // uniform branch
        const float* xrow = X + (size_t)rA * K;

        v8f acc[NT];
        #pragma unroll
        for (int t = 0; t < NT; ++t)
            #pragma unroll
            for (int j = 0; j < 8; ++j) acc[t][j] = 0.0f;

        for (int k = 0; k < K; k += 4){
            int kk = k + half*2;
            v2f a = *(const v2f*)(xrow + kk);           // A 16x4: lane=M, {K=kk, kk+1}
            const float* bp = wlds + (kk >> 1)*PSTR + ln*2;
            #pragma unroll
            for (int t = 0; t < NT; ++t){
                v2f b = *(const v2f*)(bp + t*32);       // B 4x16: one ds_load_b64
                acc[t] = __builtin_amdgcn_wmma_f32_16x16x4_f32(
                    false, a, false, b, (short)0, acc[t], false, false);
            }
        }

        #pragma unroll
        for (int t = 0; t < NT; ++t){
            float bv = bias ? bias[t*16 + ln] : 0.0f;
            #pragma unroll
            for (int v = 0; v < 8; ++v){
                int r = row0 + half*8 + v;              // C/D: vgpr v -> row v (+8 hi half)
                if (r < M) Y[(size_t)r*NC + t*16 + ln] = acc[t][v] + bv;
            }
        }
    }
}

// in-place row-wise log_softmax over 64 columns; one wave per row, 2 cols/lane
__global__ void logsoftmax_kernel(float* __restrict__ out, int n){
    int t = blockIdx.x*blockDim.x + threadIdx.x;
    int row = t >> 5;
    if (row >= n) return;
    int lane = t & 31;
    float* p = out + (size_t)row * OUTC;
    float a = p[lane], b = p[lane + 32];
    float m = fmaxf(a, b);
    #pragma unroll
    for (int o = 16; o >= 1; o >>= 1) m = fmaxf(m, __shfl_xor(m, o, 32));
    float s = expf(a - m) + expf(b - m);
    #pragma unroll
    for (int o = 16; o >= 1; o >>= 1) s += __shfl_xor(s, o, 32);
    float ls = m + logf(s);
    p[lane] = a - ls; p[lane + 32] = b - ls;
}

// ---------------- host orchestration ----------------

extern "C" void kernel_launch(void* const* d_in, const int* in_sizes, int n_in,
                              void* d_out, int out_size, void* d_ws, size_t ws_size,
                              hipStream_t stream) {
    const float* x    = (const float*)d_in[0];
    const int*   ei   = (const int*)  d_in[1];
    const int*   eib  = (const int*)  d_in[2];
    const float* lam  = (const float*)d_in[3];
    const int*   perm = (const int*)  d_in[4];
    const float* W0   = (const float*)d_in[5];
    const float* b0   = (const float*)d_in[6];
    const float* W1   = (const float*)d_in[7];
    const float* b1   = (const float*)d_in[8];
    const float* W2   = (const float*)d_in[9];
    const float* b2   = (const float*)d_in[10];
    const float* Wlin = (const float*)d_in[11];
    const float* blin = (const float*)d_in[12];
    float* out = (float*)d_out;

    const int *src = ei,  *dst  = ei  + EE;
    const int *srcb = eib, *dstb = eib + EE;

    // workspace carve-out (~129 MB total -> L2-resident on MI455X's 192 MB L2)
    size_t off = 0;
    auto alloc = [&](size_t bytes)->float*{
        float* p = (float*)((char*)d_ws + off);
        off += (bytes + 255) & ~(size_t)255;
        return p;
    };
    float* dinv    = alloc((size_t)NN*4);
    float* deginv  = alloc((size_t)NN*4);
    float* dinvb   = alloc((size_t)NN*4);
    float* deginvb = alloc((size_t)NN*4);
    const size_t NF = (size_t)NN * CH;
    float* x0   = alloc(NF*4);   // node features (reused as x_out accumulator at the end)
    float* xmix = alloc(NF*4);
    float* Ybuf = alloc(NF*4);   // x0 @ W
    float* Ymix = alloc(NF*4);   // xmix @ W
    float* agg  = alloc(NF*4);   // scatter target
    (void)ws_size; (void)in_sizes; (void)n_in; (void)out_size;

    dim3 blk(256);
    auto G = [](long long n){ return dim3((unsigned)((n + 255) / 256)); };

    // degrees & norms (deg = 1 + in-degree; dinv = deg^-1/2; deginv = deg^-1)
    fill_kernel<<<G(NN), blk, 0, stream>>>(dinv, 1.0f, NN);
    fill_kernel<<<G(NN), blk, 0, stream>>>(dinvb, 1.0f, NN);
    count_deg_kernel<<<G(EE), blk, 0, stream>>>(dst,  dinv,  EE);
    count_deg_kernel<<<G(EE), blk, 0, stream>>>(dstb, dinvb, EE);
    norms_kernel<<<G(NN), blk, 0, stream>>>(dinv,  deginv,  NN);
    norms_kernel<<<G(NN), blk, 0, stream>>>(dinvb, deginvb, NN);

    init_mix_kernel<<<G((long long)NN*32), blk, 0, stream>>>(x, perm, lam, x0, xmix, NN);

    const float* Ws[3] = {W0, W1, W2};
    const float* bs[3] = {b0, b1, b2};
    const int gemmBlocks = (NN + 127) / 128;
    const size_t lds8 = (size_t)(CH/2) * (2*CH   + 32) * 4;   // 73,728 B (<< 320 KB WGP LDS)
    const size_t lds4 = (size_t)(CH/2) * (2*OUTC + 32) * 4;   // 40,960 B

    for (int l = 0; l < 3; ++l){
        const bool last = (l == 2);
        // GEMM-first reordering: Y = x0@W shared by h-conv and x0-update; Yb = Y[perm].
        // Both GEMMs share one LDS staging of W (dual-source kernel).
        gemm_wmma_f32<8><<<gemmBlocks, blk, lds8, stream>>>(x0, Ybuf, xmix, Ymix, Ws[l], nullptr, NN);

        // main-branch aggregation A @ Y
        fill_kernel<<<G((long long)NF), blk, 0, stream>>>(agg, 0.0f, (int)NF);
        scatter_kernel<<<G((long long)EE*32), blk, 0, stream>>>(Ybuf, src, dst, nullptr, dinv, agg, EE);
        if (!last)
            // xmix <- lam*h ; x0 <- relu(agg + deginv*Y + b)  (Ybuf stays intact)
            combineA_kernel<<<G((long long)NN*32), blk, 0, stream>>>(agg, Ybuf, Ymix, deginv, bs[l], lam, xmix, x0, NN);
        else
            // x0 reused as x_out accumulator: x0 <- lam*h
            combineA_kernel<<<G((long long)NN*32), blk, 0, stream>>>(agg, Ybuf, Ymix, deginv, bs[l], lam, x0, nullptr, NN);

        // branch aggregation A_b @ Y[perm]  (gather index composed into the scatter)
        fill_kernel<<<G((long long)NF), blk, 0, stream>>>(agg, 0.0f, (int)NF);
        scatter_kernel<<<G((long long)EE*32), blk, 0, stream>>>(Ybuf, srcb, dstb, perm, dinvb, agg, EE);
        combineB_kernel<<<G((long long)NN*32), blk, 0, stream>>>(agg, Ymix, deginvb, bs[l], lam,
                                                                 last ? x0 : xmix, NN);
    }

    // logits = x_out @ Wlin + blin ; then in-place log_softmax
    gemm_wmma_f32<4><<<gemmBlocks, blk, lds4, stream>>>(x0, out, nullptr, nullptr, Wlin, blin, NN);
    logsoftmax_kernel<<<G((long long)NN*32), blk, 0, stream>>>(out, NN);
}